// GNN_Pushover_84920093376946
// MI455X (gfx1250) — compile-verified
//
#include <hip/hip_runtime.h>
#include <hip/hip_bf16.h>
#include <math.h>

#define N_NODES 100000
#define N_EDGES 1600000
#define N_GRAPH 128
#define DIM_IN  16
#define HIDC    32
#define HEADS   4
#define EDIM    8
#define OUTD    4
#define HC      128   // HEADS*HIDC

typedef float v2f __attribute__((ext_vector_type(2)));
typedef float v8f __attribute__((ext_vector_type(8)));

// float atomic max via sign-split (init pattern 0xFFFFFFFF always loses both branches)
__device__ inline void atomicMaxF(float* addr, float v) {
  int bits = __float_as_int(v);
  if (bits >= 0) atomicMax((int*)addr, bits);
  else           atomicMin((unsigned int*)addr, (unsigned int)bits);
}

__device__ inline float elu1(float x) { return x > 0.f ? x : (expf(x) - 1.f); }

// ---------------- stage 0: mean incoming edge_attr per node (self-loop attr) ----------------
__global__ void k_count_sum(const int* __restrict__ dst, const float* __restrict__ eattr,
                            float* __restrict__ loop_sum, float* __restrict__ cnt) {
  long tid = blockIdx.x * (long)blockDim.x + threadIdx.x;
  if (tid >= (long)N_EDGES * EDIM) return;
  int e = (int)(tid >> 3), j = (int)(tid & 7);
  int d = dst[e];
  atomicAdd(&loop_sum[(size_t)d * EDIM + j], eattr[tid]);
  if (j == 0) atomicAdd(&cnt[d], 1.0f);
}

__global__ void k_loop_div(float* __restrict__ loop, const float* __restrict__ cnt) {
  int tid = blockIdx.x * blockDim.x + threadIdx.x;
  if (tid >= N_NODES * EDIM) return;
  loop[tid] /= fmaxf(cnt[tid >> 3], 1.0f);
}

// ---------------- f32 WMMA GEMM: Y[N,ncols] = X[N,K] @ W[K,ncols] + bias ----------------
// one wave per 16x16 output tile; blockDim = 32*(ncols/16); grid.x = N/16
__global__ void k_gemm_bias(const float* __restrict__ X, const float* __restrict__ W,
                            const float* __restrict__ bias, float* __restrict__ Y,
                            int K, int ncols) {
  int wave = threadIdx.x >> 5;
  int lane = threadIdx.x & 31;
  int g = lane >> 4;           // lane half selects K sub-pair
  int l = lane & 15;           // row (A) / col (B,C,D)
  int row0 = blockIdx.x * 16;
  int col0 = wave * 16;
  const float* xp = X + (size_t)(row0 + l) * K + 2 * g;
  v8f acc = {0.f, 0.f, 0.f, 0.f, 0.f, 0.f, 0.f, 0.f};
  for (int k0 = 0; k0 < K; k0 += 4) {
    v2f a, b;
    a.x = xp[k0];
    a.y = xp[k0 + 1];
    const float* wp = W + (size_t)(k0 + 2 * g) * ncols + col0 + l;
    b.x = wp[0];
    b.y = wp[ncols];
    acc = __builtin_amdgcn_wmma_f32_16x16x4_f32(false, a, false, b, (short)0, acc, false, false);
  }
  float bv = bias[col0 + l];
  float* yp = Y + (size_t)row0 * ncols + col0 + l;
#pragma unroll
  for (int r = 0; r < 8; ++r)
    yp[(size_t)(r + 8 * g) * ncols] = acc[r] + bv;
}

// ---------------- edge pass 1: logits + segment max ----------------
template <int H, int C>
__global__ void k_logit_max(const int* __restrict__ src, const int* __restrict__ dst,
                            const float* __restrict__ eattr, const float* __restrict__ loop_attr,
                            const float* __restrict__ xl, const float* __restrict__ xr,
                            const float* __restrict__ we, const float* __restrict__ att,
                            float* __restrict__ logit, float* __restrict__ m) {
  const int HCl = H * C;
  __shared__ float sWe[EDIM * HC];
  __shared__ float sAtt[HC];
  for (int i = threadIdx.x; i < EDIM * HCl; i += blockDim.x) sWe[i] = we[i];
  for (int i = threadIdx.x; i < HCl; i += blockDim.x) sAtt[i] = att[i];
  __syncthreads();
  long tid = blockIdx.x * (long)blockDim.x + threadIdx.x;
  const long total = (long)(N_EDGES + N_NODES) * H;
  if (tid >= total) return;
  long e = tid / H;
  int h = (int)(tid - e * H);
  int si, di;
  float eav[EDIM];
  if (e < N_EDGES) {
    si = src[e]; di = dst[e];
#pragma unroll
    for (int j = 0; j < EDIM; ++j) eav[j] = eattr[e * EDIM + j];
  } else {
    si = di = (int)(e - N_EDGES);
#pragma unroll
    for (int j = 0; j < EDIM; ++j) eav[j] = loop_attr[(size_t)si * EDIM + j];
  }
  const float4* pl4 = (const float4*)(xl + (size_t)si * HCl + h * C);
  const float4* pr4 = (const float4*)(xr + (size_t)di * HCl + h * C);
  const float* pw = sWe + h * C;
  const float* pa = sAtt + h * C;
  float acc = 0.f;
#pragma unroll
  for (int q = 0; q < C / 4; ++q) {
    float4 vl = pl4[q];
    float4 vr = pr4[q];
    float lv[4] = {vl.x, vl.y, vl.z, vl.w};
    float rv[4] = {vr.x, vr.y, vr.z, vr.w};
#pragma unroll
    for (int u = 0; u < 4; ++u) {
      int c = q * 4 + u;
      float ea = 0.f;
#pragma unroll
      for (int j = 0; j < EDIM; ++j) ea = fmaf(eav[j], pw[j * HCl + c], ea);
      float t = lv[u] + rv[u] + ea;
      t = t > 0.f ? t : 0.2f * t;   // leaky_relu(0.2)
      acc = fmaf(pa[c], t, acc);
    }
  }
  logit[tid] = acc;
  atomicMaxF(&m[(size_t)di * H + h], acc);
}

// ---------------- edge pass 2: p=exp(logit-m); s += p; out[dst] += p*xl[src] ----------------
template <int H, int C>
__global__ void k_exp_agg(const int* __restrict__ src, const int* __restrict__ dst,
                          const float* __restrict__ logit, const float* __restrict__ m,
                          float* __restrict__ s, const float* __restrict__ xl,
                          float* __restrict__ out) {
  const int HCl = H * C;
  long tid = blockIdx.x * (long)blockDim.x + threadIdx.x;
  const long total = (long)(N_EDGES + N_NODES) * H;
  if (tid >= total) return;
  long e = tid / H;
  int h = (int)(tid - e * H);
  int si, di;
  if (e < N_EDGES) { si = src[e]; di = dst[e]; }
  else             { si = di = (int)(e - N_EDGES); }
  float p = expf(logit[tid] - m[(size_t)di * H + h]);
  atomicAdd(&s[(size_t)di * H + h], p);
  const float4* pl4 = (const float4*)(xl + (size_t)si * HCl + h * C);
  float* po = out + (size_t)di * HCl + h * C;
#pragma unroll
  for (int q = 0; q < C / 4; ++q) {
    float4 v = pl4[q];
    atomicAdd(&po[q * 4 + 0], p * v.x);
    atomicAdd(&po[q * 4 + 1], p * v.y);
    atomicAdd(&po[q * 4 + 2], p * v.z);
    atomicAdd(&po[q * 4 + 3], p * v.w);
  }
}

// ---------------- normalize + bias + elu ----------------
template <int H, int C>
__global__ void k_finalize(float* __restrict__ out, const float* __restrict__ s,
                           const float* __restrict__ bias) {
  const int HCl = H * C;
  long tid = blockIdx.x * (long)blockDim.x + threadIdx.x;
  if (tid >= (long)N_NODES * HCl) return;
  int n = (int)(tid / HCl);
  int hc = (int)(tid - (long)n * HCl);
  int h = hc / C;
  float v = out[tid] / (s[(size_t)n * H + h] + 1e-16f) + bias[hc];
  out[tid] = elu1(v);
}

// ---------------- global mean pool ----------------
__global__ void k_pool(const float* __restrict__ h3, const int* __restrict__ batch,
                       float* __restrict__ gsum, float* __restrict__ gcnt) {
  long tid = blockIdx.x * (long)blockDim.x + threadIdx.x;
  if (tid >= (long)N_NODES * HIDC) return;
  int n = (int)(tid / HIDC), c = (int)(tid - (long)n * HIDC);
  int b = batch[n];
  atomicAdd(&gsum[b * HIDC + c], h3[tid]);
  if (c == 0) atomicAdd(&gcnt[b], 1.0f);
}

// ---------------- MLP head: one thread per graph ----------------
__global__ void k_fc(const float* __restrict__ gsum, const float* __restrict__ gcnt,
                     const float* __restrict__ w1, const float* __restrict__ b1,
                     const float* __restrict__ w2, const float* __restrict__ b2,
                     float* __restrict__ out) {
  int gi = blockIdx.x * blockDim.x + threadIdx.x;
  if (gi >= N_GRAPH) return;
  float inv = 1.0f / fmaxf(gcnt[gi], 1.0f);
  float g[HIDC];
#pragma unroll
  for (int c = 0; c < HIDC; ++c) g[c] = gsum[gi * HIDC + c] * inv;
  float t[2 * HIDC];
  for (int j = 0; j < 2 * HIDC; ++j) {
    float a = b1[j];
#pragma unroll
    for (int c = 0; c < HIDC; ++c) a = fmaf(g[c], w1[c * (2 * HIDC) + j], a);
    t[j] = elu1(a);
  }
  for (int o = 0; o < OUTD; ++o) {
    float a = b2[o];
#pragma unroll
    for (int j = 0; j < 2 * HIDC; ++j) a = fmaf(t[j], w2[j * OUTD + o], a);
    out[gi * OUTD + o] = a;
  }
}

// ---------------- host-side layer driver ----------------
template <int H, int C>
static void run_layer(const float* X, int K,
                      const float* wl, const float* bl, const float* wr, const float* br,
                      const float* we, const float* att, const float* bc,
                      float* xl, float* xr, float* out, float* logit, float* m, float* sbuf,
                      const int* srcI, const int* dstI, const float* eattr,
                      const float* loop_attr, hipStream_t stream) {
  const int HCl = H * C;
  dim3 gb(N_NODES / 16);                 // 6250, exact
  int bt = 32 * (HCl / 16);              // 8 waves (HCl=128) or 2 waves (HCl=32)
  k_gemm_bias<<<gb, bt, 0, stream>>>(X, wl, bl, xl, K, HCl);
  k_gemm_bias<<<gb, bt, 0, stream>>>(X, wr, br, xr, K, HCl);
  hipMemsetAsync(m, 0xFF, (size_t)N_NODES * H * sizeof(float), stream);   // loses every max
  hipMemsetAsync(sbuf, 0, (size_t)N_NODES * H * sizeof(float), stream);
  hipMemsetAsync(out, 0, (size_t)N_NODES * HCl * sizeof(float), stream);
  long totE = (long)(N_EDGES + N_NODES) * H;
  int eb = (int)((totE + 255) / 256);
  k_logit_max<H, C><<<eb, 256, 0, stream>>>(srcI, dstI, eattr, loop_attr, xl, xr, we, att, logit, m);
  k_exp_agg<H, C><<<eb, 256, 0, stream>>>(srcI, dstI, logit, m, sbuf, xl, out);
  long totN = (long)N_NODES * HCl;
  k_finalize<H, C><<<(int)((totN + 255) / 256), 256, 0, stream>>>(out, sbuf, bc);
}

extern "C" void kernel_launch(void* const* d_in, const int* in_sizes, int n_in,
                              void* d_out, int out_size, void* d_ws, size_t ws_size,
                              hipStream_t stream) {
  (void)in_sizes; (void)n_in; (void)out_size; (void)ws_size;
  const float* x     = (const float*)d_in[0];
  const int*   ei    = (const int*)d_in[1];
  const float* eattr = (const float*)d_in[2];
  const int*   batch = (const int*)d_in[3];
  const float *wl1=(const float*)d_in[4],  *bl1=(const float*)d_in[5],
              *wr1=(const float*)d_in[6],  *br1=(const float*)d_in[7],
              *we1=(const float*)d_in[8],  *att1=(const float*)d_in[9],  *bc1=(const float*)d_in[10];
  const float *wl2=(const float*)d_in[11], *bl2=(const float*)d_in[12],
              *wr2=(const float*)d_in[13], *br2=(const float*)d_in[14],
              *we2=(const float*)d_in[15], *att2=(const float*)d_in[16], *bc2=(const float*)d_in[17];
  const float *wl3=(const float*)d_in[18], *bl3=(const float*)d_in[19],
              *wr3=(const float*)d_in[20], *br3=(const float*)d_in[21],
              *we3=(const float*)d_in[22], *att3=(const float*)d_in[23], *bc3=(const float*)d_in[24];
  const float *wfc1=(const float*)d_in[25], *bfc1=(const float*)d_in[26],
              *wfc2=(const float*)d_in[27], *bfc2=(const float*)d_in[28];
  const int* srcI = ei;
  const int* dstI = ei + N_EDGES;
  float* outp = (float*)d_out;

  float* ws = (float*)d_ws;
  size_t off = 0;
  float* xl        = ws + off; off += (size_t)N_NODES * HC;
  float* xr        = ws + off; off += (size_t)N_NODES * HC;
  float* buf       = ws + off; off += (size_t)N_NODES * HC;     // layer in/out (out reuses input slot)
  float* logit     = ws + off; off += (size_t)(N_EDGES + N_NODES) * HEADS;
  float* m         = ws + off; off += (size_t)N_NODES * HEADS;
  float* sbuf      = ws + off; off += (size_t)N_NODES * HEADS;
  float* loop_attr = ws + off; off += (size_t)N_NODES * EDIM;
  float* cnt       = ws + off; off += (size_t)N_NODES;
  float* gsum      = ws + off; off += (size_t)N_GRAPH * HIDC;
  float* gcnt      = ws + off; off += (size_t)N_GRAPH;

  // stage 0: self-loop edge_attr = mean of incoming edge_attr
  hipMemsetAsync(loop_attr, 0, (size_t)N_NODES * EDIM * sizeof(float), stream);
  hipMemsetAsync(cnt, 0, (size_t)N_NODES * sizeof(float), stream);
  {
    long t = (long)N_EDGES * EDIM;
    k_count_sum<<<(int)((t + 255) / 256), 256, 0, stream>>>(dstI, eattr, loop_attr, cnt);
    k_loop_div<<<(N_NODES * EDIM + 255) / 256, 256, 0, stream>>>(loop_attr, cnt);
  }

  // conv1: IN=16 -> 4x32 ; conv2: 128 -> 4x32 ; conv3: 128 -> 1x32
  run_layer<HEADS, HIDC>(x,   DIM_IN, wl1, bl1, wr1, br1, we1, att1, bc1,
                         xl, xr, buf, logit, m, sbuf, srcI, dstI, eattr, loop_attr, stream);
  run_layer<HEADS, HIDC>(buf, HC,     wl2, bl2, wr2, br2, we2, att2, bc2,
                         xl, xr, buf, logit, m, sbuf, srcI, dstI, eattr, loop_attr, stream);
  run_layer<1, HIDC>(buf, HC,         wl3, bl3, wr3, br3, we3, att3, bc3,
                     xl, xr, buf, logit, m, sbuf, srcI, dstI, eattr, loop_attr, stream);

  // global mean pool + MLP head
  hipMemsetAsync(gsum, 0, (size_t)N_GRAPH * HIDC * sizeof(float), stream);
  hipMemsetAsync(gcnt, 0, (size_t)N_GRAPH * sizeof(float), stream);
  {
    long t = (long)N_NODES * HIDC;
    k_pool<<<(int)((t + 255) / 256), 256, 0, stream>>>(buf, batch, gsum, gcnt);
  }
  k_fc<<<1, N_GRAPH, 0, stream>>>(gsum, gcnt, wfc1, bfc1, wfc2, bfc2, outp);
}